// WCT_58909771432284
// MI455X (gfx1250) — compile-verified
//
#include <hip/hip_runtime.h>
#include <math.h>

// Problem constants from the reference: B=4, C=512, H=W=128
#define B_ 4
#define C_ 512
#define N_ 16384
#define EPS_ 1e-5f
#define NS_ITERS 12

typedef __attribute__((ext_vector_type(2))) float v2f;
typedef __attribute__((ext_vector_type(8))) float v8f;

__device__ __forceinline__ v8f wmma4(v2f a, v2f b, v8f c) {
  // D = A(16x4) * B(4x16) + C(16x16), fp32
  return __builtin_amdgcn_wmma_f32_16x16x4_f32(false, a, false, b, (short)0, c,
                                               false, false);
}

// ---------------------------------------------------------------------------
// Per-(b,c) mean over N
// ---------------------------------------------------------------------------
__global__ void mean_kernel(const float* __restrict__ x, float* __restrict__ mean) {
  const int bc = blockIdx.x;                  // 0 .. B*C-1
  const float* p = x + (size_t)bc * N_;
  float s = 0.f;
  for (int i = threadIdx.x; i < N_; i += 256) s += p[i];
  __shared__ float red[256];
  red[threadIdx.x] = s;
  __syncthreads();
  for (int o = 128; o > 0; o >>= 1) {
    if ((int)threadIdx.x < o) red[threadIdx.x] += red[threadIdx.x + o];
    __syncthreads();
  }
  if (threadIdx.x == 0) mean[bc] = red[0] * (1.0f / (float)N_);
}

// ---------------------------------------------------------------------------
// Covariance: cov = (X*X^T - N*mu*mu^T)/(N-1) + eps*I   (mean folded in epilogue)
// One wave computes a 64x64 tile of the 512x512 output; K = 16384.
// ---------------------------------------------------------------------------
__global__ __launch_bounds__(32) void cov_kernel(const float* __restrict__ x,
                                                 const float* __restrict__ mean,
                                                 float* __restrict__ cov) {
  const int b    = blockIdx.z;
  const int row0 = blockIdx.y * 64;
  const int col0 = blockIdx.x * 64;
  const float* X  = x + (size_t)b * C_ * N_;
  const float* mu = mean + b * C_;
  float* D = cov + (size_t)b * C_ * C_;

  const int lane = threadIdx.x & 31;
  const int lm = lane & 15;   // row/col within 16
  const int kh = lane >> 4;   // K-half select

  const v8f zero = {0.f, 0.f, 0.f, 0.f, 0.f, 0.f, 0.f, 0.f};
  v8f acc[4][4];
#pragma unroll
  for (int m = 0; m < 4; ++m)
#pragma unroll
    for (int n = 0; n < 4; ++n) acc[m][n] = zero;

  const float* ap[4];
  const float* bp[4];
#pragma unroll
  for (int m = 0; m < 4; ++m) ap[m] = X + (size_t)(row0 + 16 * m + lm) * N_ + 2 * kh;
#pragma unroll
  for (int n = 0; n < 4; ++n) bp[n] = X + (size_t)(col0 + 16 * n + lm) * N_ + 2 * kh;

  for (int k = 0; k < N_; k += 4) {
    v2f av[4], bv[4];
#pragma unroll
    for (int m = 0; m < 4; ++m) av[m] = *(const v2f*)(ap[m] + k);
#pragma unroll
    for (int n = 0; n < 4; ++n) bv[n] = *(const v2f*)(bp[n] + k);
#pragma unroll
    for (int m = 0; m < 4; ++m)
#pragma unroll
      for (int n = 0; n < 4; ++n) acc[m][n] = wmma4(av[m], bv[n], acc[m][n]);
  }

  const float fac = 1.0f / (float)(N_ - 1);
#pragma unroll
  for (int m = 0; m < 4; ++m) {
#pragma unroll
    for (int n = 0; n < 4; ++n) {
      const int cidx = col0 + 16 * n + lm;
      const float mj = mu[cidx];
#pragma unroll
      for (int v = 0; v < 8; ++v) {
        const int r = row0 + 16 * m + v + 8 * kh;
        float val = (acc[m][n][v] - (float)N_ * mu[r] * mj) * fac;
        if (r == cidx) val += EPS_;
        D[(size_t)r * C_ + cidx] = val;
      }
    }
  }
}

// ---------------------------------------------------------------------------
// Trace of each 512x512 matrix (Newton-Schulz normalizer, >= lambda_max for SPD)
// ---------------------------------------------------------------------------
__global__ void trace_kernel(const float* __restrict__ cov, float* __restrict__ tr) {
  const int m = blockIdx.x;
  const float* A = cov + (size_t)m * C_ * C_;
  float s = 0.f;
  for (int i = threadIdx.x; i < C_; i += 256) s += A[(size_t)i * C_ + i];
  __shared__ float red[256];
  red[threadIdx.x] = s;
  __syncthreads();
  for (int o = 128; o > 0; o >>= 1) {
    if ((int)threadIdx.x < o) red[threadIdx.x] += red[threadIdx.x + o];
    __syncthreads();
  }
  if (threadIdx.x == 0) tr[m] = red[0];
}

// ---------------------------------------------------------------------------
// Newton-Schulz init: Y = A/tr, Z = I   (8 matrices)
// ---------------------------------------------------------------------------
__global__ void ns_init_kernel(const float* __restrict__ cov,
                               const float* __restrict__ tr,
                               float* __restrict__ Y, float* __restrict__ Z) {
  const size_t i = (size_t)blockIdx.x * 256 + threadIdx.x;
  const size_t total = (size_t)8 * C_ * C_;
  if (i >= total) return;
  const int m = (int)(i / ((size_t)C_ * C_));
  const size_t rr = i % ((size_t)C_ * C_);
  Y[i] = cov[i] / tr[m];
  Z[i] = ((rr / C_) == (rr % C_)) ? 1.f : 0.f;
}

// ---------------------------------------------------------------------------
// Batched 512x512x512 GEMM: D = alpha*(A*B) + diagBeta*I   (8 matrices)
// ---------------------------------------------------------------------------
__global__ __launch_bounds__(32) void gemm512_kernel(const float* __restrict__ Amat,
                                                     const float* __restrict__ Bmat,
                                                     float* __restrict__ Dmat,
                                                     float alpha, float diagBeta) {
  const int mm = blockIdx.z;
  const float* A  = Amat + (size_t)mm * C_ * C_;
  const float* Bp = Bmat + (size_t)mm * C_ * C_;
  float* D = Dmat + (size_t)mm * C_ * C_;
  const int row0 = blockIdx.y * 64;
  const int col0 = blockIdx.x * 64;
  const int lane = threadIdx.x & 31;
  const int lm = lane & 15;
  const int kh = lane >> 4;

  const v8f zero = {0.f, 0.f, 0.f, 0.f, 0.f, 0.f, 0.f, 0.f};
  v8f acc[4][4];
#pragma unroll
  for (int m = 0; m < 4; ++m)
#pragma unroll
    for (int n = 0; n < 4; ++n) acc[m][n] = zero;

  for (int k = 0; k < C_; k += 4) {
    v2f av[4], bv[4];
#pragma unroll
    for (int m = 0; m < 4; ++m)
      av[m] = *(const v2f*)(A + (size_t)(row0 + 16 * m + lm) * C_ + k + 2 * kh);
#pragma unroll
    for (int n = 0; n < 4; ++n) {
      const float* q = Bp + (size_t)(k + 2 * kh) * C_ + col0 + 16 * n + lm;
      v2f t;
      t.x = q[0];
      t.y = q[C_];
      bv[n] = t;
    }
#pragma unroll
    for (int m = 0; m < 4; ++m)
#pragma unroll
      for (int n = 0; n < 4; ++n) acc[m][n] = wmma4(av[m], bv[n], acc[m][n]);
  }

#pragma unroll
  for (int m = 0; m < 4; ++m)
#pragma unroll
    for (int n = 0; n < 4; ++n) {
      const int cidx = col0 + 16 * n + lm;
#pragma unroll
      for (int v = 0; v < 8; ++v) {
        const int r = row0 + 16 * m + v + 8 * kh;
        float val = alpha * acc[m][n][v];
        if (r == cidx) val += diagBeta;
        D[(size_t)r * C_ + cidx] = val;
      }
    }
}

// ---------------------------------------------------------------------------
// Whitening bias: bias[b,i] = -rsqrt(tr_c[b]) * dot(Z_b[i,:], c_mean[b,:])
// ---------------------------------------------------------------------------
__global__ void bias_white_kernel(const float* __restrict__ Z,
                                  const float* __restrict__ cmean,
                                  const float* __restrict__ tr,
                                  float* __restrict__ bias) {
  const int idx = blockIdx.x * 256 + threadIdx.x;
  if (idx >= B_ * C_) return;
  const int b = idx / C_;
  const int i = idx % C_;
  const float* r = Z + ((size_t)b * C_ + i) * C_;
  const float* mu = cmean + b * C_;
  float s = 0.f;
  for (int j = 0; j < C_; ++j) s += r[j] * mu[j];
  bias[idx] = -s * rsqrtf(tr[b]);
}

// ---------------------------------------------------------------------------
// Apply transform: out[b,i,n] = scale_b * sum_j M_b[i,j]*X[b,j,n] + bias[b,i]
// M is 512x512 per batch; X is 512x16384 per batch. 64x64 tile per wave.
// ---------------------------------------------------------------------------
__global__ __launch_bounds__(32) void apply_kernel(const float* __restrict__ Mmat,
                                                   const float* __restrict__ X,
                                                   const float* __restrict__ bias,
                                                   const float* __restrict__ tr,
                                                   int trOff, int useSqrt,
                                                   float* __restrict__ out) {
  const int b = blockIdx.z;
  const float* Mb = Mmat + (size_t)b * C_ * C_;
  const float* Xb = X + (size_t)b * C_ * N_;
  float* Ob = out + (size_t)b * C_ * N_;
  const float* bi = bias + b * C_;
  const float t = tr[trOff + b];
  const float scale = useSqrt ? sqrtf(t) : rsqrtf(t);

  const int row0 = blockIdx.y * 64;   // channel rows
  const int col0 = blockIdx.x * 64;   // pixel cols
  const int lane = threadIdx.x & 31;
  const int lm = lane & 15;
  const int kh = lane >> 4;

  const v8f zero = {0.f, 0.f, 0.f, 0.f, 0.f, 0.f, 0.f, 0.f};
  v8f acc[4][4];
#pragma unroll
  for (int m = 0; m < 4; ++m)
#pragma unroll
    for (int n = 0; n < 4; ++n) acc[m][n] = zero;

  for (int k = 0; k < C_; k += 4) {
    v2f av[4], bv[4];
#pragma unroll
    for (int m = 0; m < 4; ++m)
      av[m] = *(const v2f*)(Mb + (size_t)(row0 + 16 * m + lm) * C_ + k + 2 * kh);
#pragma unroll
    for (int n = 0; n < 4; ++n) {
      const float* q = Xb + (size_t)(k + 2 * kh) * N_ + col0 + 16 * n + lm;
      v2f tt;
      tt.x = q[0];
      tt.y = q[N_];
      bv[n] = tt;
    }
#pragma unroll
    for (int m = 0; m < 4; ++m)
#pragma unroll
      for (int n = 0; n < 4; ++n) acc[m][n] = wmma4(av[m], bv[n], acc[m][n]);
  }

#pragma unroll
  for (int m = 0; m < 4; ++m)
#pragma unroll
    for (int n = 0; n < 4; ++n) {
      const int c = col0 + 16 * n + lm;
#pragma unroll
      for (int v = 0; v < 8; ++v) {
        const int r = row0 + 16 * m + v + 8 * kh;
        Ob[(size_t)r * N_ + c] = scale * acc[m][n][v] + bi[r];
      }
    }
}

// ---------------------------------------------------------------------------
extern "C" void kernel_launch(void* const* d_in, const int* in_sizes, int n_in,
                              void* d_out, int out_size, void* d_ws, size_t ws_size,
                              hipStream_t stream) {
  (void)in_sizes; (void)n_in; (void)out_size; (void)ws_size;
  const float* content = (const float*)d_in[0];
  const float* style   = (const float*)d_in[1];
  float* out = (float*)d_out;
  float* ws  = (float*)d_ws;

  const size_t CM = (size_t)C_ * C_;   // 262144 floats per matrix

  // Workspace layout (floats)
  float* c_mean  = ws;                     // B*C
  float* s_mean  = c_mean + B_ * C_;       // B*C
  float* tr      = s_mean + B_ * C_;       // 8
  float* bias_w  = tr + 8;                 // B*C
  float* cov_all = bias_w + B_ * C_;       // 8*CM (0..3 content, 4..7 style)
  float* Y  = cov_all + 8 * CM;            // 8*CM
  float* Z  = Y + 8 * CM;                  // 8*CM
  float* Y2 = Z + 8 * CM;                  // 8*CM
  float* Z2 = Y2 + 8 * CM;                 // 8*CM
  float* T  = Z2 + 8 * CM;                 // 8*CM
  float* whit = T + 8 * CM;                // B*C*N

  // 1) means
  mean_kernel<<<B_ * C_, 256, 0, stream>>>(content, c_mean);
  mean_kernel<<<B_ * C_, 256, 0, stream>>>(style, s_mean);

  // 2) covariances (WMMA Gram + mean/eps epilogue)
  dim3 covGrid(C_ / 64, C_ / 64, B_);      // (8,8,4), 1 wave per block
  cov_kernel<<<covGrid, 32, 0, stream>>>(content, c_mean, cov_all);
  cov_kernel<<<covGrid, 32, 0, stream>>>(style, s_mean, cov_all + 4 * CM);

  // 3) traces for normalization
  trace_kernel<<<8, 256, 0, stream>>>(cov_all, tr);

  // 4) Newton-Schulz: Y -> (A/tr)^{1/2}, Z -> (A/tr)^{-1/2}, 8 matrices batched
  {
    const size_t tot = 8 * CM;
    ns_init_kernel<<<(unsigned)((tot + 255) / 256), 256, 0, stream>>>(cov_all, tr, Y, Z);
  }
  dim3 g512(C_ / 64, C_ / 64, 8);          // (8,8,8)
  float* y = Y; float* z = Z; float* y2 = Y2; float* z2 = Z2;
  for (int it = 0; it < NS_ITERS; ++it) {
    gemm512_kernel<<<g512, 32, 0, stream>>>(z, y, T, -0.5f, 1.5f);  // T = 1.5I - 0.5 Z Y
    gemm512_kernel<<<g512, 32, 0, stream>>>(y, T, y2, 1.0f, 0.0f);  // Y' = Y T
    gemm512_kernel<<<g512, 32, 0, stream>>>(T, z, z2, 1.0f, 0.0f);  // Z' = T Z
    float* tmp;
    tmp = y; y = y2; y2 = tmp;
    tmp = z; z = z2; z2 = tmp;
  }
  // Now: content cov^{-1/2} = rsqrt(tr[b]) * z[b]        (b = 0..3)
  //      style   cov^{+1/2} = sqrt(tr[4+b]) * y[4+b]

  // 5) whitening bias: -rsqrt(tr_c) * Z * c_mean
  bias_white_kernel<<<(B_ * C_ + 255) / 256, 256, 0, stream>>>(z, c_mean, tr, bias_w);

  // 6) whitened = rsqrt(tr_c)*(Z_c @ content) + bias_w   (mean subtraction folded)
  dim3 gApply(N_ / 64, C_ / 64, B_);       // (256, 8, 4)
  apply_kernel<<<gApply, 32, 0, stream>>>(z, content, bias_w, tr, 0, 0, whit);

  // 7) out = sqrt(tr_s)*(Y_s @ whitened) + s_mean
  apply_kernel<<<gApply, 32, 0, stream>>>(y + 4 * CM, whit, s_mean, tr, 4, 1, out);
}